// Decoder_71657234366479
// MI455X (gfx1250) — compile-verified
//
#include <hip/hip_runtime.h>
#include <hip/hip_bf16.h>
#include <math.h>

#define N_NODES 4096
#define K_DEG   32
#define EPS     1e-5f

typedef float v2f __attribute__((ext_vector_type(2)));
typedef float v8f __attribute__((ext_vector_type(8)));

// D = A(16x4,f32) * B(4x16,f32) + C, one wave32 per tile.
static __device__ inline v8f wmma_f32_k4(v2f a, v2f b, v8f c) {
    // 8-arg form: (neg_a, A, neg_b, B, c_mod, C, reuse_a, reuse_b)
    return __builtin_amdgcn_wmma_f32_16x16x4_f32(false, a, false, b, (short)0, c,
                                                 false, false);
}

// ---------------------------------------------------------------------------
// Y[4096, DOUT] = s ⊙ (X[4096,DIN] @ Wᵀ) + bias    (one 16x16 tile per wave)
//   MODEB==0 : W = tw,        s = BN scale (or 1),  bias = 0          -> A'
//   MODEB==1 : W = pw - tw,   s = BN scale (or 1),
//              bias = s*(tb+pb) + (beta - mean*s)                     -> B'
// All shape/mode parameters are compile-time so the K-loop is a pure
// load + v_wmma stream with no per-iteration control flow.
// ---------------------------------------------------------------------------
template <int DIN, int DOUT, int MODEB, bool BN>
__global__ void edgeconv_gemm(const float* __restrict__ X,
                              const float* __restrict__ tw, const float* __restrict__ pw,
                              const float* __restrict__ tb, const float* __restrict__ pb,
                              const float* __restrict__ gamma, const float* __restrict__ beta,
                              const float* __restrict__ mean, const float* __restrict__ var,
                              float* __restrict__ Y)
{
    constexpr int NTX = (DOUT + 15) / 16;
    constexpr int NTM = N_NODES / 16;              // 256 row tiles

    const int wave = (blockIdx.x * blockDim.x + threadIdx.x) >> 5;
    const int lane = threadIdx.x & 31;
    if (wave >= NTM * NTX) return;                 // wave-uniform guard

    const int tm   = wave / NTX;
    const int tn   = wave - tm * NTX;
    const int row0 = tm * 16;
    const int col0 = tn * 16;
    const int l15  = lane & 15;
    const int kh   = lane >> 4;                    // K half: lanes16-31 -> K+2

    // Per-lane output column (fixed across all 8 accum elements).
    const int  n      = col0 + l15;
    const bool nvalid = (n < DOUT);
    const int  nc     = nvalid ? n : 0;            // clamped for weight loads

    // Fold BN + biases into an affine epilogue (exact: applied before the max).
    float s = 1.0f, bias = 0.0f;
    if (BN) s = gamma[nc] * rsqrtf(var[nc] + EPS);
    if (MODEB) {
        float t = BN ? (beta[nc] - mean[nc] * s) : 0.0f;
        bias = s * (tb[nc] + pb[nc]) + t;
    }

    const float* xrow = X  + (size_t)(row0 + l15) * DIN + 2 * kh;  // A: M=l15
    const float* wt   = tw + (size_t)nc * DIN + 2 * kh;            // B: N=l15
    const float* wp   = pw + (size_t)nc * DIN + 2 * kh;

    v8f acc = {};
    #pragma unroll
    for (int k = 0; k < DIN; k += 4) {             // DIN ∈ {16,32,64,128}
        v2f a = *(const v2f*)(xrow + k);
        v2f b = *(const v2f*)(wt + k);
        if (MODEB) {
            v2f bp = *(const v2f*)(wp + k);
            b = bp - b;                            // (pw - tw) on the fly
        }
        acc = wmma_f32_k4(a, b, acc);
    }

    if (nvalid) {
        #pragma unroll
        for (int r = 0; r < 8; ++r) {              // v[r]: row r+8*kh, col n
            int row = row0 + r + 8 * kh;
            Y[(size_t)row * DOUT + n] = s * acc[r] + bias;
        }
    }
}

// ---------------------------------------------------------------------------
// out[v][c] = B'[v][c] + max_{j<32} A'[src[32v+j]][c]
// (node-major edge list: dst = repeat(arange(N), 32) -> neighbors contiguous)
// ---------------------------------------------------------------------------
template <int DOUT>
__global__ void gather_max(const float* __restrict__ A, const float* __restrict__ B,
                           const int* __restrict__ src, float* __restrict__ out)
{
    const int tid = blockIdx.x * blockDim.x + threadIdx.x;
    if (tid >= N_NODES * DOUT) return;
    const int v = tid / DOUT;                      // constant divisor
    const int c = tid - v * DOUT;
    const int* sv = src + v * K_DEG;
    float m = -INFINITY;
    #pragma unroll 8
    for (int j = 0; j < K_DEG; ++j) {
        int u = sv[j];
        m = fmaxf(m, A[(size_t)u * DOUT + c]);     // coalesced across channels
    }
    out[tid] = B[tid] + m;
}

// ---------------------------------------------------------------------------
// g[i][j][c] = e_c[i] · e_c[j],  c=0..2, e_c: [4096,8]
// One 16x16 tile per wave, 2 WMMAs per channel (K=8), non-temporal interleaved
// stores (write-once 201 MB stream; keep L2 for the gathers/e-buffers).
// ---------------------------------------------------------------------------
static __device__ inline v8f gram_tile(const float* __restrict__ e,
                                       int arow, int brow, int kh)
{
    v8f acc = {};
    v2f a0 = *(const v2f*)(e + arow * 8 + 2 * kh);
    v2f b0 = *(const v2f*)(e + brow * 8 + 2 * kh);
    acc = wmma_f32_k4(a0, b0, acc);
    v2f a1 = *(const v2f*)(e + arow * 8 + 4 + 2 * kh);
    v2f b1 = *(const v2f*)(e + brow * 8 + 4 + 2 * kh);
    acc = wmma_f32_k4(a1, b1, acc);
    return acc;
}

__global__ void gram3(const float* __restrict__ e1, const float* __restrict__ e2,
                      const float* __restrict__ e3, float* __restrict__ g)
{
    const int wave = (blockIdx.x * blockDim.x + threadIdx.x) >> 5;
    const int lane = threadIdx.x & 31;
    constexpr int NT = N_NODES / 16;               // 256 tiles per dim
    if (wave >= NT * NT) return;
    const int ti = wave / NT;
    const int tj = wave - ti * NT;
    const int l15 = lane & 15;
    const int kh  = lane >> 4;
    const int arow = ti * 16 + l15;
    const int brow = tj * 16 + l15;

    v8f d0 = gram_tile(e1, arow, brow, kh);
    v8f d1 = gram_tile(e2, arow, brow, kh);
    v8f d2 = gram_tile(e3, arow, brow, kh);

    const int j = tj * 16 + l15;                   // this lane's column
    #pragma unroll
    for (int r = 0; r < 8; ++r) {
        int i = ti * 16 + r + 8 * kh;
        size_t off = ((size_t)i * N_NODES + j) * 3;   // [i][j][c] channel-last
        __builtin_nontemporal_store(d0[r], g + off + 0);
        __builtin_nontemporal_store(d1[r], g + off + 1);
        __builtin_nontemporal_store(d2[r], g + off + 2);
    }
}

// ---------------------------------------------------------------------------
extern "C" void kernel_launch(void* const* d_in, const int* in_sizes, int n_in,
                              void* d_out, int out_size, void* d_ws, size_t ws_size,
                              hipStream_t stream)
{
    const float* x   = (const float*)d_in[0];
    const int*   src = (const int*)d_in[1];
    // d_in[2] = dst (implicit: dst[e] = e / 32, not needed)

    float* w    = (float*)d_ws;
    float* bufA = w;                           // 4096*128
    float* bufB = bufA + N_NODES * 128;
    float* h1   = bufB + N_NODES * 128;        // ping
    float* h2   = h1   + N_NODES * 128;        // pong
    float* hs   = h2   + N_NODES * 128;        // shared features [4096,128]
    float* eb1  = hs   + N_NODES * 128;        // [4096,8]
    float* eb2  = eb1  + N_NODES * 8;
    float* eb3  = eb2  + N_NODES * 8;

    float* out_n = (float*)d_out;              // [4096,7]
    float* out_g = out_n + N_NODES * 7;        // [4096,4096,3]

    auto run_layer = [&](auto din_c, auto dout_c, auto bn_c, const float* X,
                         int base, float* Yout) {
        constexpr int DIN  = decltype(din_c)::value;
        constexpr int DOUT = decltype(dout_c)::value;
        constexpr bool BN  = decltype(bn_c)::value;
        const float* tw = (const float*)d_in[base + 0];
        const float* tb = (const float*)d_in[base + 1];
        const float* pw = (const float*)d_in[base + 2];
        const float* pb = (const float*)d_in[base + 3];
        const float* ga = BN ? (const float*)d_in[base + 4] : nullptr;
        const float* be = BN ? (const float*)d_in[base + 5] : nullptr;
        const float* me = BN ? (const float*)d_in[base + 6] : nullptr;
        const float* va = BN ? (const float*)d_in[base + 7] : nullptr;
        constexpr int NTX = (DOUT + 15) / 16;
        const int nblocks = 32 * NTX;          // 8 waves/block, 256 row tiles
        edgeconv_gemm<DIN, DOUT, 0, BN><<<nblocks, 256, 0, stream>>>(
            X, tw, pw, tb, pb, ga, be, me, va, bufA);
        edgeconv_gemm<DIN, DOUT, 1, BN><<<nblocks, 256, 0, stream>>>(
            X, tw, pw, tb, pb, ga, be, me, va, bufB);
        constexpr int TOTAL = N_NODES * DOUT;
        gather_max<DOUT><<<(TOTAL + 255) / 256, 256, 0, stream>>>(bufA, bufB,
                                                                  src, Yout);
    };

    using std::integral_constant;
    auto c7   = integral_constant<int, 7>{};
    auto c8   = integral_constant<int, 8>{};
    auto c16  = integral_constant<int, 16>{};
    auto c32  = integral_constant<int, 32>{};
    auto c64  = integral_constant<int, 64>{};
    auto c128 = integral_constant<int, 128>{};
    auto bnT  = integral_constant<bool, true>{};
    auto bnF  = integral_constant<bool, false>{};

    // Param pointer bases (insertion order: tw,tb,pw,pb[,gamma,beta,mean,var]).
    run_layer(c64,  c32,  bnT, x,  3,  h1);    // shared[0]
    run_layer(c32,  c64,  bnT, h1, 11, h2);    // shared[1]
    run_layer(c64,  c128, bnT, h2, 19, hs);    // shared[2]
    run_layer(c128, c7,   bnF, hs, 27, out_n); // node head -> d_out[0:4096*7]
    run_layer(c128, c32,  bnT, hs, 31, h1);    // e1
    run_layer(c32,  c16,  bnT, h1, 39, h2);
    run_layer(c16,  c8,   bnT, h2, 47, eb1);
    run_layer(c128, c32,  bnT, hs, 55, h1);    // e2
    run_layer(c32,  c16,  bnT, h1, 63, h2);
    run_layer(c16,  c8,   bnT, h2, 71, eb2);
    run_layer(c128, c32,  bnT, hs, 79, h1);    // e3
    run_layer(c32,  c16,  bnT, h1, 87, h2);
    run_layer(c16,  c8,   bnT, h2, 95, eb3);

    // 65536 tiles, 8 waves per 256-thread block.
    gram3<<<8192, 256, 0, stream>>>(eb1, eb2, eb3, out_g);
}